// EstimatorModel_61864708931785
// MI455X (gfx1250) — compile-verified
//
#include <hip/hip_runtime.h>

#define NLA 512
#define NPA 2048
#define NTOT 2560
#define LF 16
#define PF 21
#define JF 16
#define HM 16
#define NL_G 64
#define NP_G 256
#define HSTR 20          // padded h row stride (17 used + 3 zero pad)
#define CUT2 25.0f

typedef float v2f __attribute__((ext_vector_type(2)));
typedef float v8f __attribute__((ext_vector_type(8)));

__device__ __forceinline__ float silu_f(float x) {
    return x / (1.0f + expf(-x));
}

__device__ __forceinline__ v8f wmma_f32_k4(v2f a, v2f b, v8f c) {
    // V_WMMA_F32_16X16X4_F32 : D = A(16x4) * B(4x16) + C(16x16), exact fp32
    return __builtin_amdgcn_wmma_f32_16x16x4_f32(false, a, false, b, (short)0, c, false, false);
}

// ---------------------------------------------------------------------------
// Kernel 1: per-node MLP embeddings -> h (NTOT x HSTR), col16 = t, 17..19 = 0
// ---------------------------------------------------------------------------
__global__ void k_embed(const float* __restrict__ xl, const float* __restrict__ xp,
                        const float* __restrict__ t, const int* __restrict__ mask,
                        const float* __restrict__ aW1, const float* __restrict__ ab1,
                        const float* __restrict__ aW2, const float* __restrict__ ab2,
                        const float* __restrict__ fW1, const float* __restrict__ fb1,
                        const float* __restrict__ fW2, const float* __restrict__ fb2,
                        const float* __restrict__ rW1, const float* __restrict__ rb1,
                        const float* __restrict__ rW2, const float* __restrict__ rb2,
                        float* __restrict__ h) {
    int node = blockIdx.x * blockDim.x + threadIdx.x;
    if (node >= NTOT) return;
    float out16[JF];
    if (node < NLA) {
        const float* W1 = mask[node] ? aW1 : fW1;
        const float* b1 = mask[node] ? ab1 : fb1;
        const float* W2 = mask[node] ? aW2 : fW2;
        const float* b2 = mask[node] ? ab2 : fb2;
        const float* x = xl + node * LF;
        float z[2 * LF];
        for (int c = 0; c < 2 * LF; c++) {
            float s = b1[c];
            for (int k = 0; k < LF; k++) s += x[k] * W1[k * (2 * LF) + c];
            z[c] = silu_f(s);
        }
        for (int c = 0; c < JF; c++) {
            float s = b2[c];
            for (int k = 0; k < 2 * LF; k++) s += z[k] * W2[k * JF + c];
            out16[c] = s;
        }
    } else {
        int p = node - NLA;
        const float* x = xp + p * PF;
        float z[2 * PF];
        for (int c = 0; c < 2 * PF; c++) {
            float s = rb1[c];
            for (int k = 0; k < PF; k++) s += x[k] * rW1[k * (2 * PF) + c];
            z[c] = silu_f(s);
        }
        for (int c = 0; c < JF; c++) {
            float s = rb2[c];
            for (int k = 0; k < 2 * PF; k++) s += z[k] * rW2[k * JF + c];
            out16[c] = s;
        }
    }
    float* hr = h + node * HSTR;
    for (int c = 0; c < JF; c++) hr[c] = out16[c];
    hr[16] = t[0];
    hr[17] = 0.0f; hr[18] = 0.0f; hr[19] = 0.0f;
}

// ---------------------------------------------------------------------------
// Kernel 2: WMMA fp32 GEMM: hA = h @ We[0:17] + be ; hB = h @ We[17:34]
// One wave per 16-row M tile (160 waves). K padded 17 -> 20 via h's zero cols.
// ---------------------------------------------------------------------------
__global__ void k_proj(const float* __restrict__ h, const float* __restrict__ We,
                       const float* __restrict__ be,
                       float* __restrict__ hA, float* __restrict__ hB) {
    int wave = (blockIdx.x * blockDim.x + threadIdx.x) >> 5;
    int lane = threadIdx.x & 31;
    if (wave >= NTOT / 16) return;           // wave-uniform
    int half = lane >> 4;
    int rc = lane & 15;
    int row0 = wave * 16;
    v8f accA = {0,0,0,0,0,0,0,0};
    v8f accB = {0,0,0,0,0,0,0,0};
    for (int kb = 0; kb < HSTR; kb += 4) {
        int k0 = kb + half * 2;
        v2f a, bA, bB;
        a.x = h[(row0 + rc) * HSTR + k0];
        a.y = h[(row0 + rc) * HSTR + k0 + 1];
        bA.x = (k0     < 17) ? We[(k0    ) * HM + rc] : 0.0f;
        bA.y = (k0 + 1 < 17) ? We[(k0 + 1) * HM + rc] : 0.0f;
        bB.x = (k0     < 17) ? We[(17 + k0    ) * HM + rc] : 0.0f;
        bB.y = (k0 + 1 < 17) ? We[(17 + k0 + 1) * HM + rc] : 0.0f;
        accA = wmma_f32_k4(a, bA, accA);
        accB = wmma_f32_k4(a, bB, accB);
    }
    for (int v = 0; v < 8; v++) {
        int rr = row0 + v + half * 8;
        hA[rr * HM + rc] = accA[v] + be[rc];   // fold be into hA
        hB[rr * HM + rc] = accB[v];
    }
}

// ---------------------------------------------------------------------------
// Kernel 3: message pass. One wave32 per ligand atom; lanes stride the 320
// same-batch neighbors (64 ligand + 256 protein), shuffle-tree reduction.
// ---------------------------------------------------------------------------
__global__ void k_pairs(const float* __restrict__ posL, const float* __restrict__ posP,
                        const float* __restrict__ hA, const float* __restrict__ hB,
                        const float* __restrict__ We, const float* __restrict__ Wx,
                        const float* __restrict__ bx,
                        float* __restrict__ aggO, float* __restrict__ transO) {
    int wave = (blockIdx.x * blockDim.x + threadIdx.x) >> 5;
    int lane = threadIdx.x & 31;
    if (wave >= NLA) return;                 // wave-uniform
    int i = wave;
    int b = i / NL_G;
    float pix = posL[i * 3 + 0], piy = posL[i * 3 + 1], piz = posL[i * 3 + 2];
    float hAi[HM], wc[HM], wx[HM];
    for (int k = 0; k < HM; k++) {
        hAi[k] = hA[i * HM + k];             // already includes be
        wc[k]  = We[34 * HM + k];
        wx[k]  = Wx[k];
    }
    float bxv = bx[0];
    float agg[HM];
    for (int k = 0; k < HM; k++) agg[k] = 0.0f;
    float tx = 0.0f, ty = 0.0f, tz = 0.0f;
    for (int it = 0; it < (NL_G + NP_G) / 32; it++) {
        int jl = lane + it * 32;
        int j; const float* pj;
        if (jl < NL_G) { j = b * NL_G + jl; pj = posL + j * 3; }
        else { int pq = b * NP_G + (jl - NL_G); j = NLA + pq; pj = posP + pq * 3; }
        float dx = pix - pj[0], dy = piy - pj[1], dz = piz - pj[2];
        float d2 = dx * dx + dy * dy + dz * dz;
        float g = (d2 < CUT2) ? 1.0f : 0.0f;
        const float4* hb4 = (const float4*)(hB + j * HM);
        float4 h0 = hb4[0], h1 = hb4[1], h2 = hb4[2], h3 = hb4[3];
        float hbj[HM] = {h0.x, h0.y, h0.z, h0.w, h1.x, h1.y, h1.z, h1.w,
                         h2.x, h2.y, h2.z, h2.w, h3.x, h3.y, h3.z, h3.w};
        float em = 0.0f;
        for (int k = 0; k < HM; k++) {
            float m = silu_f(hAi[k] + hbj[k] + d2 * wc[k]);
            agg[k] += g * m;
            em += m * wx[k];
        }
        float e = g * (em + bxv);
        tx += e * dx; ty += e * dy; tz += e * dz;
    }
    for (int off = 16; off > 0; off >>= 1) {
        for (int k = 0; k < HM; k++) agg[k] += __shfl_down(agg[k], off, 32);
        tx += __shfl_down(tx, off, 32);
        ty += __shfl_down(ty, off, 32);
        tz += __shfl_down(tz, off, 32);
    }
    if (lane == 0) {
        for (int k = 0; k < HM; k++) aggO[i * HM + k] = agg[k];
        transO[i * 4 + 0] = tx;
        transO[i * 4 + 1] = ty;
        transO[i * 4 + 2] = tz;
        transO[i * 4 + 3] = 0.0f;
    }
}

// ---------------------------------------------------------------------------
// Kernel 4: tail GEMM chain via WMMA, one wave per 16-atom tile (32 tiles).
//   h_out = [h|agg](16x33) @ Wh + bh
//   y     = silu(h_out @ decW1 + db1)        (16x32)
//   x_dec = y @ decW2 + db2                  (16x16)
//   out[r] = mask ? [x_dec | trans] : 0      (512x19)
// ---------------------------------------------------------------------------
__device__ __forceinline__ float loadA1(const float* h, const float* agg, int row, int k) {
    if (k < 17) return h[row * HSTR + k];
    if (k < 33) return agg[row * HM + (k - 17)];
    return 0.0f;
}

__global__ void k_tail(const float* __restrict__ h, const float* __restrict__ agg,
                       const float* __restrict__ trans, const int* __restrict__ mask,
                       const float* __restrict__ Wh, const float* __restrict__ bh,
                       const float* __restrict__ dW1, const float* __restrict__ db1,
                       const float* __restrict__ dW2, const float* __restrict__ db2,
                       float* __restrict__ out) {
    __shared__ float sm_h[16 * 16];
    __shared__ float sm_y[16 * 32];
    int tile = blockIdx.x;
    int lane = threadIdx.x & 31;
    int half = lane >> 4;
    int rc = lane & 15;
    int row0 = tile * 16;

    // GEMM1: [h|agg] @ Wh, K=33 padded to 36
    v8f acc = {0,0,0,0,0,0,0,0};
    for (int kb = 0; kb < 36; kb += 4) {
        int k0 = kb + half * 2;
        v2f a, bm;
        a.x = loadA1(h, agg, row0 + rc, k0);
        a.y = loadA1(h, agg, row0 + rc, k0 + 1);
        bm.x = (k0     < 33) ? Wh[(k0    ) * JF + rc] : 0.0f;
        bm.y = (k0 + 1 < 33) ? Wh[(k0 + 1) * JF + rc] : 0.0f;
        acc = wmma_f32_k4(a, bm, acc);
    }
    for (int v = 0; v < 8; v++) sm_h[(v + half * 8) * 16 + rc] = acc[v] + bh[rc];
    __syncthreads();

    // GEMM2: sm_h(16x16) @ dW1(16x32), silu epilogue
    v8f acc0 = {0,0,0,0,0,0,0,0};
    v8f acc1 = {0,0,0,0,0,0,0,0};
    for (int kb = 0; kb < 16; kb += 4) {
        int k0 = kb + half * 2;
        v2f a, b0, b1;
        a.x = sm_h[rc * 16 + k0];
        a.y = sm_h[rc * 16 + k0 + 1];
        b0.x = dW1[(k0    ) * 32 + rc];      b0.y = dW1[(k0 + 1) * 32 + rc];
        b1.x = dW1[(k0    ) * 32 + 16 + rc]; b1.y = dW1[(k0 + 1) * 32 + 16 + rc];
        acc0 = wmma_f32_k4(a, b0, acc0);
        acc1 = wmma_f32_k4(a, b1, acc1);
    }
    for (int v = 0; v < 8; v++) {
        int r = v + half * 8;
        sm_y[r * 32 + rc]      = silu_f(acc0[v] + db1[rc]);
        sm_y[r * 32 + 16 + rc] = silu_f(acc1[v] + db1[16 + rc]);
    }
    __syncthreads();

    // GEMM3: sm_y(16x32) @ dW2(32x16)
    v8f acc2 = {0,0,0,0,0,0,0,0};
    for (int kb = 0; kb < 32; kb += 4) {
        int k0 = kb + half * 2;
        v2f a, bm;
        a.x = sm_y[rc * 32 + k0];
        a.y = sm_y[rc * 32 + k0 + 1];
        bm.x = dW2[(k0    ) * JF + rc];
        bm.y = dW2[(k0 + 1) * JF + rc];
        acc2 = wmma_f32_k4(a, bm, acc2);
    }
    for (int v = 0; v < 8; v++) {
        int r = row0 + v + half * 8;
        float val = acc2[v] + db2[rc];
        out[r * 19 + rc] = mask[r] ? val : 0.0f;
    }
    // trans columns 16..18
    if (lane < 16) {
        int r = row0 + lane;
        float mm = mask[r] ? 1.0f : 0.0f;
        out[r * 19 + 16] = mm * trans[r * 4 + 0];
        out[r * 19 + 17] = mm * trans[r * 4 + 1];
        out[r * 19 + 18] = mm * trans[r * 4 + 2];
    }
}

// ---------------------------------------------------------------------------
extern "C" void kernel_launch(void* const* d_in, const int* in_sizes, int n_in,
                              void* d_out, int out_size, void* d_ws, size_t ws_size,
                              hipStream_t stream) {
    (void)in_sizes; (void)n_in; (void)out_size; (void)ws_size;
    const float* xl   = (const float*)d_in[0];
    const float* xp   = (const float*)d_in[1];
    const float* posL = (const float*)d_in[2];
    const float* posP = (const float*)d_in[3];
    const float* t    = (const float*)d_in[4];
    // d_in[5], d_in[6] (batch arrays) are implicit in the index math
    const int*   mask = (const int*)d_in[7];
    const float* aW1 = (const float*)d_in[8],  *ab1 = (const float*)d_in[9];
    const float* aW2 = (const float*)d_in[10], *ab2 = (const float*)d_in[11];
    const float* fW1 = (const float*)d_in[12], *fb1 = (const float*)d_in[13];
    const float* fW2 = (const float*)d_in[14], *fb2 = (const float*)d_in[15];
    const float* rW1 = (const float*)d_in[16], *rb1 = (const float*)d_in[17];
    const float* rW2 = (const float*)d_in[18], *rb2 = (const float*)d_in[19];
    const float* dW1 = (const float*)d_in[20], *db1 = (const float*)d_in[21];
    const float* dW2 = (const float*)d_in[22], *db2 = (const float*)d_in[23];
    const float* We  = (const float*)d_in[24], *be  = (const float*)d_in[25];
    const float* Wx  = (const float*)d_in[26], *bx  = (const float*)d_in[27];
    const float* Wh  = (const float*)d_in[28], *bh  = (const float*)d_in[29];
    float* out = (float*)d_out;

    float* ws = (float*)d_ws;
    float* h     = ws;                                   // 2560*20
    float* hA    = h + NTOT * HSTR;                      // 2560*16
    float* hB    = hA + NTOT * HM;                       // 2560*16
    float* aggW  = hB + NTOT * HM;                       // 512*16
    float* trW   = aggW + NLA * HM;                      // 512*4

    k_embed<<<dim3((NTOT + 255) / 256), dim3(256), 0, stream>>>(
        xl, xp, t, mask, aW1, ab1, aW2, ab2, fW1, fb1, fW2, fb2,
        rW1, rb1, rW2, rb2, h);

    k_proj<<<dim3((NTOT / 16) / 4), dim3(128), 0, stream>>>(h, We, be, hA, hB);

    k_pairs<<<dim3(NLA / 8), dim3(256), 0, stream>>>(
        posL, posP, hA, hB, We, Wx, bx, aggW, trW);

    k_tail<<<dim3(NLA / 16), dim3(32), 0, stream>>>(
        h, aggW, trW, mask, Wh, bh, dW1, db1, dW2, db2, out);
}